// deformableconv_49314814493261
// MI455X (gfx1250) — compile-verified
//
#include <hip/hip_runtime.h>
#include <math.h>

// Modulated deformable conv (DCNv2) for MI455X / gfx1250, wave32.
// Kernel 1: offset(18ch)+modulator(9ch) 3x3 conv as WMMA f32 GEMM -> d_ws.
// Kernel 2: fused bilinear deformable sampling + K=576 GEMM (Cout=64) using
//           V_WMMA_F32_16X16X4_F32 accumulation.
// Workspace use: 4*18*65536 + 4*9*65536 floats = ~28.3 MB in d_ws.

typedef __attribute__((ext_vector_type(2))) float v2f;
typedef __attribute__((ext_vector_type(8))) float v8f;

#define HH   256
#define WW   256
#define HW   65536
#define CIN  64
#define COUT 64
#define NOFF 18
#define NMOD 9
#define NB   4

__device__ __forceinline__ v8f wmma_f32(v2f a, v2f b, v8f c) {
  // 8 args: (neg_a, A, neg_b, B, c_mod, C, reuse_a, reuse_b)
  return __builtin_amdgcn_wmma_f32_16x16x4_f32(
      false, a, false, b, (short)0, c, false, false);
}

// ---------------------------------------------------------------------------
// Kernel 1: offset/modulator conv.  Wave tile: M=16 pixels x N=32 (27 real).
// ---------------------------------------------------------------------------
__global__ __launch_bounds__(256) void offmod_conv_kernel(
    const float* __restrict__ x, const float* __restrict__ offw,
    const float* __restrict__ offb, const float* __restrict__ modw,
    const float* __restrict__ modb, float* __restrict__ off_out,
    float* __restrict__ mod_out) {
  __shared__ float wt[64 * 32];        // weights for current tap: [c][n]
  __shared__ float vals[8 * 16 * 64];  // per-wave im2col slice: [m][c]

  const int tid  = threadIdx.x;
  const int lane = tid & 31;
  const int wave = tid >> 5;
  const int mloc = lane & 15;  // pixel-in-tile (A rows) and N-in-tile (B cols)
  const int lh   = lane >> 4;  // lane half -> K sub-pair
  const int p0   = blockIdx.x * 128 + wave * 16;
  const int p    = p0 + mloc;
  const int bn   = p >> 16;         // batch (HW = 65536)
  const int rem  = p & (HW - 1);
  const int h    = rem >> 8;
  const int w    = rem & 255;

  float* myval = &vals[wave * 16 * 64];

  v8f acc0 = {};
  v8f acc1 = {};

  for (int kk = 0; kk < 9; ++kk) {
    __syncthreads();
    // Stage weight slice Wt[c][n] = {offset_w | mod_w}[n][c][kk], n padded to 32
    for (int i = tid; i < 64 * 32; i += 256) {
      const int c = i >> 5, n = i & 31;
      float wv = 0.f;
      if (n < NOFF)                wv = offw[(n * CIN + c) * 9 + kk];
      else if (n < NOFF + NMOD)    wv = modw[((n - NOFF) * CIN + c) * 9 + kk];
      wt[i] = wv;
    }
    // Stage A: regular (non-deformed) 3x3 tap with zero padding
    const int di = kk / 3 - 1, dj = kk % 3 - 1;
    const int yy = h + di, xx = w + dj;
    const bool inb = (yy >= 0) && (yy < HH) && (xx >= 0) && (xx < WW);
    const int idx = inb ? (yy * WW + xx) : 0;
    const float* xp = x + (size_t)(bn * CIN + lh * 32) * HW;
    for (int c = 0; c < 32; ++c) {
      myval[mloc * 64 + lh * 32 + c] = inb ? xp[idx] : 0.f;
      xp += HW;
    }
    __syncthreads();
    // GEMM over this tap's 64 channels, K-step = 4
    for (int c4 = 0; c4 < 64; c4 += 4) {
      const v2f a = *(const v2f*)&myval[mloc * 64 + c4 + 2 * lh];
      v2f b0, b1;
      b0.x = wt[(c4 + 2 * lh) * 32 + mloc];
      b0.y = wt[(c4 + 2 * lh + 1) * 32 + mloc];
      b1.x = wt[(c4 + 2 * lh) * 32 + 16 + mloc];
      b1.y = wt[(c4 + 2 * lh + 1) * 32 + 16 + mloc];
      acc0 = wmma_f32(a, b0, acc0);
      acc1 = wmma_f32(a, b1, acc1);
    }
  }

  // Epilogue. C/D layout: VGPR j -> M = j + 8*lh, N = mloc (+16 for acc1).
  const int pb    = p0 >> 16;
  const int rem0  = p0 & (HW - 1);
  const int rbase = rem0 + lh * 8;
  {  // N-tile 0: offset channels 0..15
    const int n = mloc;
    const float bia = offb[n];
    float* dst = off_out + (size_t)(pb * NOFF + n) * HW + rbase;
    float4 v0 = {acc0[0] + bia, acc0[1] + bia, acc0[2] + bia, acc0[3] + bia};
    float4 v1 = {acc0[4] + bia, acc0[5] + bia, acc0[6] + bia, acc0[7] + bia};
    *(float4*)(dst)     = v0;
    *(float4*)(dst + 4) = v1;
  }
  {  // N-tile 1: channels 16..31 (16,17 offsets; 18..26 modulator; rest pad)
    const int n = 16 + mloc;
    if (n < NOFF) {
      const float bia = offb[n];
      float* dst = off_out + (size_t)(pb * NOFF + n) * HW + rbase;
      float4 v0 = {acc1[0] + bia, acc1[1] + bia, acc1[2] + bia, acc1[3] + bia};
      float4 v1 = {acc1[4] + bia, acc1[5] + bia, acc1[6] + bia, acc1[7] + bia};
      *(float4*)(dst)     = v0;
      *(float4*)(dst + 4) = v1;
    } else if (n < NOFF + NMOD) {
      const int nm = n - NOFF;
      const float bia = modb[nm];
      float* dst = mod_out + (size_t)(pb * NMOD + nm) * HW + rbase;
      float4 v0, v1;
      v0.x = 2.f / (1.f + __expf(-(acc1[0] + bia)));
      v0.y = 2.f / (1.f + __expf(-(acc1[1] + bia)));
      v0.z = 2.f / (1.f + __expf(-(acc1[2] + bia)));
      v0.w = 2.f / (1.f + __expf(-(acc1[3] + bia)));
      v1.x = 2.f / (1.f + __expf(-(acc1[4] + bia)));
      v1.y = 2.f / (1.f + __expf(-(acc1[5] + bia)));
      v1.z = 2.f / (1.f + __expf(-(acc1[6] + bia)));
      v1.w = 2.f / (1.f + __expf(-(acc1[7] + bia)));
      *(float4*)(dst)     = v0;
      *(float4*)(dst + 4) = v1;
    }
  }
}

// ---------------------------------------------------------------------------
// Kernel 2: fused deformable sampling + GEMM.  Wave: M=16 pixels x N=64.
// ---------------------------------------------------------------------------
__global__ __launch_bounds__(256) void deform_gemm_kernel(
    const float* __restrict__ x, const float* __restrict__ regw,
    const float* __restrict__ off_in, const float* __restrict__ mod_in,
    float* __restrict__ out) {
  __shared__ float wt[64 * 64];        // reg_w slice for current tap: [c][n]
  __shared__ float vals[8 * 16 * 64];  // per-wave sampled slice: [m][c]

  const int tid  = threadIdx.x;
  const int lane = tid & 31;
  const int wave = tid >> 5;
  const int mloc = lane & 15;
  const int lh   = lane >> 4;
  const int p0   = blockIdx.x * 128 + wave * 16;
  const int p    = p0 + mloc;
  const int bn   = p >> 16;
  const int rem  = p & (HW - 1);
  const int h    = rem >> 8;
  const int w    = rem & 255;

  float* myval = &vals[wave * 16 * 64];

  v8f acc[4] = {v8f{}, v8f{}, v8f{}, v8f{}};

  for (int kk = 0; kk < 9; ++kk) {
    __syncthreads();
    // Stage weight slice Wt[c][n] = reg_w[n][c][kk]
    for (int i = tid; i < 64 * 64; i += 256) {
      const int c = i >> 6, n = i & 63;
      wt[i] = regw[(n * CIN + c) * 9 + kk];
    }
    // Per-pixel bilinear setup for this tap
    const float dy  = off_in[(size_t)(bn * NOFF + kk * 2 + 0) * HW + rem];
    const float dx  = off_in[(size_t)(bn * NOFF + kk * 2 + 1) * HW + rem];
    const float msk = mod_in[(size_t)(bn * NMOD + kk) * HW + rem];
    const float py = dy + (float)(kk / 3) + (float)(h - 1);
    const float px = dx + (float)(kk % 3) + (float)(w - 1);
    const float y0f = floorf(py), x0f = floorf(px);
    const float ty = py - y0f, tx = px - x0f;
    const int y0 = (int)y0f, x0 = (int)x0f;
    const int y1 = y0 + 1, x1 = x0 + 1;
    const float in00 = (y0 >= 0 && y0 < HH && x0 >= 0 && x0 < WW) ? 1.f : 0.f;
    const float in01 = (y0 >= 0 && y0 < HH && x1 >= 0 && x1 < WW) ? 1.f : 0.f;
    const float in10 = (y1 >= 0 && y1 < HH && x0 >= 0 && x0 < WW) ? 1.f : 0.f;
    const float in11 = (y1 >= 0 && y1 < HH && x1 >= 0 && x1 < WW) ? 1.f : 0.f;
    const float w00 = (1.f - ty) * (1.f - tx) * msk * in00;
    const float w01 = (1.f - ty) * tx * msk * in01;
    const float w10 = ty * (1.f - tx) * msk * in10;
    const float w11 = ty * tx * msk * in11;
    const int yc0 = min(max(y0, 0), HH - 1), yc1 = min(max(y1, 0), HH - 1);
    const int xc0 = min(max(x0, 0), WW - 1), xc1 = min(max(x1, 0), WW - 1);
    const int i00 = yc0 * WW + xc0, i01 = yc0 * WW + xc1;
    const int i10 = yc1 * WW + xc0, i11 = yc1 * WW + xc1;
    const float* xp = x + (size_t)(bn * CIN + lh * 32) * HW;
    for (int c = 0; c < 32; ++c) {
      const float v = w00 * xp[i00] + w01 * xp[i01] +
                      w10 * xp[i10] + w11 * xp[i11];
      myval[mloc * 64 + lh * 32 + c] = v;
      xp += HW;
    }
    __syncthreads();
    // GEMM over this tap's 64 channels, 4 N-tiles, K-step = 4
    for (int c4 = 0; c4 < 64; c4 += 4) {
      const v2f a = *(const v2f*)&myval[mloc * 64 + c4 + 2 * lh];
#pragma unroll
      for (int nt = 0; nt < 4; ++nt) {
        v2f bb;
        bb.x = wt[(c4 + 2 * lh) * 64 + nt * 16 + mloc];
        bb.y = wt[(c4 + 2 * lh + 1) * 64 + nt * 16 + mloc];
        acc[nt] = wmma_f32(a, bb, acc[nt]);
      }
    }
  }

  // Store: VGPR j -> pixel p0 + j + 8*lh, channel nt*16 + mloc.
  const int pb    = p0 >> 16;
  const int rem0  = p0 & (HW - 1);
  const int rbase = rem0 + lh * 8;
#pragma unroll
  for (int nt = 0; nt < 4; ++nt) {
    const int n = nt * 16 + mloc;
    float* dst = out + (size_t)(pb * COUT + n) * HW + rbase;
    float4 v0 = {acc[nt][0], acc[nt][1], acc[nt][2], acc[nt][3]};
    float4 v1 = {acc[nt][4], acc[nt][5], acc[nt][6], acc[nt][7]};
    *(float4*)(dst)     = v0;
    *(float4*)(dst + 4) = v1;
  }
}

// ---------------------------------------------------------------------------
extern "C" void kernel_launch(void* const* d_in, const int* in_sizes, int n_in,
                              void* d_out, int out_size, void* d_ws,
                              size_t ws_size, hipStream_t stream) {
  const float* x    = (const float*)d_in[0];
  const float* offw = (const float*)d_in[1];
  const float* offb = (const float*)d_in[2];
  const float* modw = (const float*)d_in[3];
  const float* modb = (const float*)d_in[4];
  const float* regw = (const float*)d_in[5];
  float* out = (float*)d_out;

  float* off_buf = (float*)d_ws;                         // 4*18*65536 floats
  float* mod_buf = off_buf + (size_t)NB * NOFF * HW;     // 4*9*65536 floats

  const int pixels = NB * HW;        // 262144
  const int blocks = pixels / 128;   // 2048 (exact)

  offmod_conv_kernel<<<blocks, 256, 0, stream>>>(x, offw, offb, modw, modb,
                                                 off_buf, mod_buf);
  deform_gemm_kernel<<<blocks, 256, 0, stream>>>(x, regw, off_buf, mod_buf,
                                                 out);
}